// LSTMStack_48954037240323
// MI455X (gfx1250) — compile-verified
//
#include <hip/hip_runtime.h>
#include <math.h>

// MI455X / gfx1250, wave32. Bandwidth-bound FP32 LSTM stack (batch=1, 1 step).
// Strategy: split-K matvec partials via V_WMMA_F32_16X16X4_F32, deterministic
// two-phase reduction, per-layer fused cell activation, then FC.

typedef __attribute__((ext_vector_type(2))) float v2f;
typedef __attribute__((ext_vector_type(8))) float v8f;

#define KSLICE 256          // K elements per split
#define KCHUNK 128          // LDS staging chunk
#define ROWS_PER_WAVE 16    // WMMA M
#define WAVES_PER_BLOCK 4
#define LDS_STRIDE 132      // 128 + 4 pad -> conflict-free ds_load_b64 frags

// Computes part[split][rowBase+r] = dot(W[row, kb:kb+256], v[kb:kb+256])
// split < K/KSLICE uses (W0, v0); otherwise (W1, v1) with rebased kb.
__global__ __launch_bounds__(128) void matvec16_partial(
    const float* __restrict__ W0, const float* __restrict__ W1,
    const float* __restrict__ v0, const float* __restrict__ v1,
    float* __restrict__ part, int rowsTotal, int K)
{
    __shared__ float hs[KSLICE];
    __shared__ float tile[WAVES_PER_BLOCK][ROWS_PER_WAVE * LDS_STRIDE];

    const int tid   = threadIdx.x;
    const int lane  = tid & 31;
    const int wv    = tid >> 5;
    const int split = blockIdx.y;
    const int rowBase = (blockIdx.x * WAVES_PER_BLOCK + wv) * ROWS_PER_WAVE;

    const int spm = K / KSLICE;   // splits per matrix (8 when K=2048)
    const float* W; const float* v; int kb;
    if (split < spm) { W = W0; v = v0; kb = split * KSLICE; }
    else             { W = W1; v = v1; kb = (split - spm) * KSLICE; }

    // Stage the shared h-slice for this split (block-wide).
    if (tid < KSLICE / 4) {
        *(float4*)&hs[tid * 4] = *(const float4*)&v[kb + tid * 4];
    }
    __syncthreads();

    const int m     = lane & 15;            // WMMA A row (M) for this lane
    const int khalf = (lane >> 4) << 1;     // lanes 0-15 -> K{0,1}; 16-31 -> K{2,3}

    v8f acc = {};   // 16x16 f32 accumulator (every column == row dot)

    for (int ch = 0; ch < KSLICE / KCHUNK; ++ch) {
        // Stage 16 rows x 128 cols, fully coalesced: 512B per instruction.
        const float* Wb = W + (size_t)rowBase * K + kb + ch * KCHUNK;
        #pragma unroll
        for (int r = 0; r < ROWS_PER_WAVE; ++r) {
            float4 t = *(const float4*)&Wb[(size_t)r * K + lane * 4];
            *(float4*)&tile[wv][r * LDS_STRIDE + lane * 4] = t;
        }
        // 32 WMMAs over this chunk (K step 4). A/B frags per ISA 7.12.2:
        //   A: VGPR0/1 = W[m][4kk+khalf], W[m][4kk+khalf+1]
        //   B: broadcast h-chunk across all 16 columns.
        #pragma unroll
        for (int kk = 0; kk < KCHUNK / 4; ++kk) {
            v2f a = *(const v2f*)&tile[wv][m * LDS_STRIDE + kk * 4 + khalf];
            v2f b = *(const v2f*)&hs[ch * KCHUNK + kk * 4 + khalf];
            acc = __builtin_amdgcn_wmma_f32_16x16x4_f32(
                false, a, false, b, (short)0, acc, false, false);
        }
    }

    // D layout: lane 0 VGPR i = row i (col 0); lane 16 VGPR i = row 8+i.
    float* dst = part + (size_t)split * rowsTotal + rowBase;
    if (lane == 0) {
        #pragma unroll
        for (int i = 0; i < 8; ++i) dst[i] = acc[i];
    } else if (lane == 16) {
        #pragma unroll
        for (int i = 0; i < 8; ++i) dst[8 + i] = acc[i];
    }
}

// Phase 2 per layer: reduce split partials, add biases, LSTM cell math.
__global__ __launch_bounds__(256) void lstm_cell_kernel(
    const float* __restrict__ part,   // [nsplit][4H]
    const float* __restrict__ b_ih, const float* __restrict__ b_hh,
    const float* __restrict__ c_prev, float* __restrict__ h_out,
    int H, int nsplit)
{
    int j = blockIdx.x * blockDim.x + threadIdx.x;
    if (j >= H) return;
    float g[4];
    #pragma unroll
    for (int gate = 0; gate < 4; ++gate) {
        int row = gate * H + j;
        float s = b_ih[row] + b_hh[row];
        for (int sp = 0; sp < nsplit; ++sp) s += part[(size_t)sp * 4 * H + row];
        g[gate] = s;
    }
    float ig = 1.f / (1.f + expf(-g[0]));
    float fg = 1.f / (1.f + expf(-g[1]));
    float gg = tanhf(g[2]);
    float og = 1.f / (1.f + expf(-g[3]));
    float c  = fg * c_prev[j] + ig * gg;
    h_out[j] = og * tanhf(c);
}

// FC reduce: out[j] = fc_b[j] + sum_sp part[sp][j]
__global__ __launch_bounds__(256) void fc_reduce_kernel(
    const float* __restrict__ part, const float* __restrict__ fc_b,
    float* __restrict__ out, int n, int nsplit)
{
    int j = blockIdx.x * blockDim.x + threadIdx.x;
    if (j >= n) return;
    float s = fc_b[j];
    for (int sp = 0; sp < nsplit; ++sp) s += part[(size_t)sp * n + j];
    out[j] = s;
}

extern "C" void kernel_launch(void* const* d_in, const int* in_sizes, int n_in,
                              void* d_out, int out_size, void* d_ws, size_t ws_size,
                              hipStream_t stream) {
    (void)in_sizes; (void)n_in; (void)out_size; (void)ws_size;
    const float* x    = (const float*)d_in[0];
    const float* h0   = (const float*)d_in[1];   // [L,1,H]
    const float* c0   = (const float*)d_in[2];   // [L,1,H]
    const float* W_ih = (const float*)d_in[3];   // [L,4H,H]
    const float* W_hh = (const float*)d_in[4];   // [L,4H,H]
    const float* b_ih = (const float*)d_in[5];   // [L,4H]
    const float* b_hh = (const float*)d_in[6];   // [L,4H]
    const float* fc_w = (const float*)d_in[7];   // [DOUT,H]
    const float* fc_b = (const float*)d_in[8];   // [DOUT]
    float* out = (float*)d_out;

    const int H = 2048, L = 4, R = 4 * H /*8192*/, DOUT = 1024;
    const int NSPLIT = 2 * (H / KSLICE);   // 16 (Wi + Wh halves)
    const int NSPLIT_FC = H / KSLICE;      // 8

    float* ws    = (float*)d_ws;
    float* part  = ws;                              // 16 * 8192
    float* hbuf  = part + (size_t)NSPLIT * R;       // L * H
    float* part2 = hbuf + (size_t)L * H;            // 8 * 1024

    dim3 mvBlock(WAVES_PER_BLOCK * 32);
    for (int l = 0; l < L; ++l) {
        const float* hin   = (l == 0) ? x : (hbuf + (size_t)(l - 1) * H);
        const float* hprev = h0 + (size_t)l * H;
        dim3 grid(R / (ROWS_PER_WAVE * WAVES_PER_BLOCK), NSPLIT);   // 128 x 16
        matvec16_partial<<<grid, mvBlock, 0, stream>>>(
            W_ih + (size_t)l * R * H, W_hh + (size_t)l * R * H,
            hin, hprev, part, R, H);
        lstm_cell_kernel<<<dim3(H / 256), dim3(256), 0, stream>>>(
            part, b_ih + (size_t)l * R, b_hh + (size_t)l * R,
            c0 + (size_t)l * H, hbuf + (size_t)l * H, H, NSPLIT);
    }
    // FC: rows=1024, K=2048, single matrix -> all splits hit (W0, v0).
    dim3 gridFC(DOUT / (ROWS_PER_WAVE * WAVES_PER_BLOCK), NSPLIT_FC); // 16 x 8
    matvec16_partial<<<gridFC, mvBlock, 0, stream>>>(
        fc_w, fc_w, hbuf + (size_t)3 * H, hbuf + (size_t)3 * H, part2, DOUT, H);
    fc_reduce_kernel<<<dim3(DOUT / 256), dim3(256), 0, stream>>>(
        part2, fc_b, out, DOUT, NSPLIT_FC);
}